// WaveletTransform_1279900254221
// MI455X (gfx1250) — compile-verified
//
#include <hip/hip_runtime.h>

// ---------------------------------------------------------------------------
// Batched 4-level db4 wavedec (pywt 'symmetric'), one workgroup per row.
// Row staged to LDS once; all 4 levels computed in LDS; coefficient bands
// written straight to HBM  ->  ~1 read + 1 write of DRAM (bandwidth bound,
// ~134 MB -> ~5.8 us floor at 23.3 TB/s).
// Filter math runs on the matrix pipe via V_WMMA_F32_16X16X4_F32:
//   C[m,c] = sum_q A_q[m,k] * B_q[k,c],  m = (phase j' 0..7) + 8*(filter f),
//   A_q = banded Toeplitz chunks of the db4 dec filters (lane-constant),
//   B_q[k,c] = ext[256t + 16c + 4q + k]  (LDS gather, ds_load_2addr_b64).
// Stores: full tiles use unconditional b128 stores (coalesced); only the
// final partial tile per level takes the predicated scalar path.
// ---------------------------------------------------------------------------

typedef __attribute__((ext_vector_type(2))) float v2f;
typedef __attribute__((ext_vector_type(4))) float v4f;
typedef __attribute__((ext_vector_type(8))) float v8f;

#define ROW_LEN   8192
#define TOT_LEN   8218
#define NTHREADS  256
#define CAP_A     8480   // >= 2 + (8192+13) + tile over-read (261), padded
#define CAP_B     4384   // >= 2 + (4099+13) + tile over-read (261)

// pywt db4 rec_lo; dec_lo = rec_lo[::-1]; correlation kernel K0 = rec_lo,
// K1[k] = (-1)^k * rec_lo[7-k]   (== dec_hi[::-1])
__constant__ float c_rec_lo[8] = {
     0.23037781330885523f,  0.7148465705525415f,   0.6308807679295904f,
    -0.02798376941698385f, -0.18703481171888114f,  0.030841381835986965f,
     0.032883011666982945f, -0.010597401784997278f };

__device__ __forceinline__ float dwt_tap(int f, int d) {
    if (d < 0 || d > 7) return 0.0f;                 // outside the band
    float lo = c_rec_lo[d];
    float hi = (d & 1) ? -c_rec_lo[7 - d] : c_rec_lo[7 - d];
    return f ? hi : lo;
}

__device__ __forceinline__ v8f wmma_f32_4(v2f a, v2f b, v8f c) {
#if __has_builtin(__builtin_amdgcn_wmma_f32_16x16x4_f32)
    return __builtin_amdgcn_wmma_f32_16x16x4_f32(false, a, false, b,
                                                 (short)0, c, false, false);
#else
    asm("v_wmma_f32_16x16x4_f32 %0, %1, %2, %0" : "+v"(c) : "v"(a), "v"(b));
    return c;
#endif
}

// One DWT level: ein = extended input (LDS), eout = next-level extended buffer
// (LDS, cA written at eout[6+J]), cd = global pointer for this level's details.
__device__ __forceinline__ void dwt_level(
    const float* __restrict__ ein, float* __restrict__ eout,
    float* __restrict__ cd, int outlen, const v2f (&Aq)[6],
    int wave, int nwaves, int lane)
{
    const int  ntiles = (outlen + 127) >> 7;   // 128 outputs (per filter)/tile
    const int  cc     = lane & 15;             // column within tile
    const bool hiHalf = lane >= 16;            // lanes 16..31 carry K=2,3 / cD
    const int  koff   = hiHalf ? 2 : 0;

    for (int t = wave; t < ntiles; t += nwaves) {   // wave-uniform: EXEC all-1
        const int ebase = (t << 8) + (cc << 4) + koff;
        v8f acc = {0.f, 0.f, 0.f, 0.f, 0.f, 0.f, 0.f, 0.f};
#pragma unroll
        for (int q = 0; q < 6; ++q) {
            const float* p = ein + ebase + (q << 2);  // even index -> 8B align
            v2f b; b.x = p[0]; b.y = p[1];
            acc = wmma_f32_4(Aq[q], b, acc);
        }
        // C VGPR g: lanes 0-15 -> cA (f=0, j'=g), lanes 16-31 -> cD (f=1).
        // Jb has low 3 bits zero -> LDS dest (buf + 8 + Jb) is 32B aligned.
        const int Jb = (t << 7) + (cc << 3);
        if (Jb + 8 <= outlen) {                 // full 8-wide slot: fast path
            v4f lo4 = {acc[0], acc[1], acc[2], acc[3]};
            v4f hi4 = {acc[4], acc[5], acc[6], acc[7]};
            if (!hiHalf) {                      // cA -> LDS, 2x ds_store_b128
                v4f* p = (v4f*)(eout + 6 + Jb);
                p[0] = lo4;
                p[1] = hi4;
            } else {                            // cD -> HBM, 16B coalesced
                __builtin_memcpy(cd + Jb,     &lo4, sizeof(v4f));
                __builtin_memcpy(cd + Jb + 4, &hi4, sizeof(v4f));
            }
        } else {                                // final partial tile only
            if (!hiHalf) {
#pragma unroll
                for (int g = 0; g < 8; ++g) {
                    int J = Jb + g;
                    if (J < outlen) eout[6 + J] = acc[g];
                }
            } else {
#pragma unroll
                for (int g = 0; g < 8; ++g) {
                    int J = Jb + g;
                    if (J < outlen) cd[J] = acc[g];
                }
            }
        }
    }
}

// Half-sample symmetric extension edges: e[i] (i<6) = x[5-i],
// e[n+6+k] = x[n-1-k]; interior already at e[6..6+n).
__device__ __forceinline__ void fix_edges(float* e, int n, int tid) {
    if (tid < 6)        e[tid]         = e[11 - tid];
    else if (tid < 13) { int k = tid - 6; e[n + 6 + k] = e[n + 5 - k]; }
}

__global__ __launch_bounds__(NTHREADS)
void db4_wavedec4_kernel(const float* __restrict__ x, float* __restrict__ out)
{
    __shared__ __attribute__((aligned(16))) float bufA[CAP_A];
    __shared__ __attribute__((aligned(16))) float bufB[CAP_B];

    const int tid    = threadIdx.x;
    const int lane   = tid & 31;
    const int wave   = tid >> 5;
    const int nwaves = NTHREADS >> 5;
    const int row    = blockIdx.x;

    // Per-lane banded-Toeplitz filter chunks (ISA 16x4 f32 A layout:
    // M = lane&15; lanes<16 hold K=0,1; lanes>=16 hold K=2,3).
    v2f Aq[6];
    {
        const int m  = lane & 15;
        const int jp = m & 7;             // output phase 0..7
        const int f  = m >> 3;            // 0 = lo (cA), 1 = hi (cD)
        const int kb = (lane < 16) ? 0 : 2;
#pragma unroll
        for (int q = 0; q < 6; ++q) {
            Aq[q].x = dwt_tap(f, 4 * q + kb     - 2 * jp);
            Aq[q].y = dwt_tap(f, 4 * q + kb + 1 - 2 * jp);
        }
    }

    // Zero LDS once (keeps tile over-reads finite).
    for (int i = tid; i < CAP_A; i += NTHREADS) bufA[i] = 0.0f;
    for (int i = tid; i < CAP_B; i += NTHREADS) bufB[i] = 0.0f;
    __syncthreads();

    float* extA = bufA + 2;   // ext[i]; interior starts at bufA+8 (32B align)
    float* extB = bufB + 2;

    // Stage this row into LDS with b128 loads/stores.
    {
        const float4* src = (const float4*)(x + (size_t)row * ROW_LEN);
        float4*       dst = (float4*)(bufA + 8);
        for (int i = tid; i < (ROW_LEN >> 2); i += NTHREADS) dst[i] = src[i];
    }
    __syncthreads();
    fix_edges(extA, ROW_LEN, tid);
    __syncthreads();

    float* gout = out + (size_t)row * TOT_LEN;

    // Output layout: [cA4 @0 (518) | cD4 @518 (518) | cD3 @1036 (1030)
    //                 | cD2 @2066 (2053) | cD1 @4119 (4099)]
    dwt_level(extA, extB, gout + 4119, 4099, Aq, wave, nwaves, lane); // L1
    __syncthreads();
    fix_edges(extB, 4099, tid);
    __syncthreads();

    dwt_level(extB, extA, gout + 2066, 2053, Aq, wave, nwaves, lane); // L2
    __syncthreads();
    fix_edges(extA, 2053, tid);
    __syncthreads();

    dwt_level(extA, extB, gout + 1036, 1030, Aq, wave, nwaves, lane); // L3
    __syncthreads();
    fix_edges(extB, 1030, tid);
    __syncthreads();

    dwt_level(extB, extA, gout + 518,  518,  Aq, wave, nwaves, lane); // L4
    __syncthreads();

    // cA4: copy from LDS to the head of the row's output.
    for (int i = tid; i < 518; i += NTHREADS) gout[i] = extA[6 + i];
}

extern "C" void kernel_launch(void* const* d_in, const int* in_sizes, int n_in,
                              void* d_out, int out_size, void* d_ws, size_t ws_size,
                              hipStream_t stream) {
    (void)n_in; (void)out_size; (void)d_ws; (void)ws_size;
    const float* x   = (const float*)d_in[0];
    float*       out = (float*)d_out;
    const int rows = in_sizes[0] / ROW_LEN;   // 64*32 = 2048 rows
    db4_wavedec4_kernel<<<rows, NTHREADS, 0, stream>>>(x, out);
}